// RecurrentGCN_EGCNH_36859409335074
// MI455X (gfx1250) — compile-verified
//
#include <hip/hip_runtime.h>
#include <math.h>

typedef float v2f __attribute__((ext_vector_type(2)));
typedef float v8f __attribute__((ext_vector_type(8)));

#define FDIM 128

__device__ __forceinline__ void atomic_add_f32(float* p, float v) {
    __hip_atomic_fetch_add(p, v, __ATOMIC_RELAXED, __HIP_MEMORY_SCOPE_AGENT);
}
__device__ __forceinline__ int atomic_add_i32(int* p, int v) {
    return __hip_atomic_fetch_add(p, v, __ATOMIC_RELAXED, __HIP_MEMORY_SCOPE_AGENT);
}

// ---------- stage 0: 1/||p|| ----------
__global__ void k_pnorm(const float* __restrict__ p, float* __restrict__ pinv) {
    __shared__ float s[FDIM];
    int t = threadIdx.x;
    float v = p[t];
    s[t] = v * v;
    __syncthreads();
    for (int st = FDIM / 2; st > 0; st >>= 1) {
        if (t < st) s[t] += s[t + st];
        __syncthreads();
    }
    if (t == 0) pinv[0] = rsqrtf(s[0]);
}

// ---------- stage 1: score[n] = tanh((x[n].p) / ||p||) ----------
__global__ void k_score(const float* __restrict__ x, const float* __restrict__ p,
                        const float* __restrict__ pinv, float* __restrict__ score, int n_nodes) {
    int n = blockIdx.x * blockDim.x + threadIdx.x;
    if (n >= n_nodes) return;
    const float4* xr = (const float4*)(x + (size_t)n * FDIM);
    const float4* pp = (const float4*)p;
    float acc = 0.f;
    #pragma unroll
    for (int i = 0; i < FDIM / 4; ++i) {
        float4 a = xr[i], b = pp[i];
        acc += a.x * b.x + a.y * b.y + a.z * b.z + a.w * b.w;
    }
    score[n] = tanhf(acc * pinv[0]);
}

// ---------- stage 2: iterative top-128 (single block) ----------
__global__ void __launch_bounds__(1024) k_topk(float* __restrict__ score, int* __restrict__ perm,
                                               float* __restrict__ topval, int n_nodes) {
    __shared__ float sval[1024];
    __shared__ int   sidx[1024];
    int tid = threadIdx.x;
    for (int k = 0; k < FDIM; ++k) {
        float best = -3.0e38f;
        int bidx = 0;
        for (int i = tid; i < n_nodes; i += 1024) {
            float v = score[i];
            if (v > best) { best = v; bidx = i; }
        }
        sval[tid] = best; sidx[tid] = bidx;
        __syncthreads();
        for (int s = 512; s > 0; s >>= 1) {
            if (tid < s && sval[tid + s] > sval[tid]) {
                sval[tid] = sval[tid + s]; sidx[tid] = sidx[tid + s];
            }
            __syncthreads();
        }
        if (tid == 0) {
            perm[k] = sidx[0];
            topval[k] = sval[0];
            score[sidx[0]] = -3.0e38f;   // remove winner
        }
        __syncthreads();
    }
}

// ---------- stage 3: x_tilde[i][j] = x[perm[i]][j] * topval[i] ----------
__global__ void k_xtilde(const float* __restrict__ x, const int* __restrict__ perm,
                         const float* __restrict__ topval, float* __restrict__ xt) {
    int t = blockIdx.x * blockDim.x + threadIdx.x;   // 128*128
    int i = t >> 7, j = t & (FDIM - 1);
    xt[t] = x[(size_t)perm[i] * FDIM + j] * topval[i];
}

// ---------- stage 4: GRU input/hidden matmuls (tiny, scalar) ----------
__global__ void k_gru_mm(const float* __restrict__ xt, const float* __restrict__ W0,
                         const float* __restrict__ w_ih, const float* __restrict__ w_hh,
                         const float* __restrict__ b_ih, const float* __restrict__ b_hh,
                         float* __restrict__ gi, float* __restrict__ gh) {
    int t = blockIdx.x * blockDim.x + threadIdx.x;   // 128*384
    if (t >= FDIM * 3 * FDIM) return;
    int i = t / (3 * FDIM), j = t % (3 * FDIM);
    const float* wi = w_ih + (size_t)j * FDIM;
    const float* wh = w_hh + (size_t)j * FDIM;
    const float* xr = xt + (size_t)i * FDIM;
    const float* hr = W0 + (size_t)i * FDIM;
    float a = 0.f, b = 0.f;
    #pragma unroll 4
    for (int k = 0; k < FDIM; ++k) { a += xr[k] * wi[k]; b += hr[k] * wh[k]; }
    gi[t] = a + b_ih[j];
    gh[t] = b + b_hh[j];
}

// ---------- stage 5: GRU gate combine -> evolved W ----------
__global__ void k_gru_combine(const float* __restrict__ gi, const float* __restrict__ gh,
                              const float* __restrict__ W0, float* __restrict__ W) {
    int t = blockIdx.x * blockDim.x + threadIdx.x;   // 128*128
    int i = t >> 7, j = t & (FDIM - 1);
    int base = i * 3 * FDIM + j;
    float r = 1.f / (1.f + expf(-(gi[base] + gh[base])));
    float z = 1.f / (1.f + expf(-(gi[base + FDIM] + gh[base + FDIM])));
    float n = tanhf(gi[base + 2 * FDIM] + r * gh[base + 2 * FDIM]);
    W[t] = (1.f - z) * n + z * W0[t];
}

// ---------- zero fill ----------
__global__ void k_zero_f32(float* __restrict__ p, int n) {
    int i = blockIdx.x * blockDim.x + threadIdx.x;
    if (i < n) p[i] = 0.f;
}
__global__ void k_zero_i32(int* __restrict__ p, int n) {
    int i = blockIdx.x * blockDim.x + threadIdx.x;
    if (i < n) p[i] = 0;
}

// ---------- stage 6a: weighted degree (edges + self loops) ----------
__global__ void k_deg(const int* __restrict__ ei, const float* __restrict__ ew,
                      float* __restrict__ deg, int E, int n_nodes) {
    int e = blockIdx.x * blockDim.x + threadIdx.x;
    if (e >= E + n_nodes) return;
    if (e < E) atomic_add_f32(&deg[ei[E + e]], ew[e]);   // col = edge_index[1][e]
    else       atomic_add_f32(&deg[e - E], 1.0f);        // self loop
}

__global__ void k_dinv(const float* __restrict__ deg, float* __restrict__ dinv, int n_nodes) {
    int i = blockIdx.x * blockDim.x + threadIdx.x;
    if (i >= n_nodes) return;
    float d = deg[i];
    dinv[i] = d > 0.f ? rsqrtf(d) : 0.f;
}

// ---------- stage 6b: CSR-by-target build ----------
__global__ void k_count(const int* __restrict__ ei, int* __restrict__ cnt, int E, int n_nodes) {
    int e = blockIdx.x * blockDim.x + threadIdx.x;
    if (e >= E + n_nodes) return;
    int col = (e < E) ? ei[E + e] : (e - E);
    atomic_add_i32(&cnt[col], 1);
}

// single-block hierarchical exclusive scan over n counts -> rowptr[0..n]
__global__ void __launch_bounds__(1024) k_scan(const int* __restrict__ cnt,
                                               int* __restrict__ rowptr, int n) {
    __shared__ int ssum[1024];
    int tid = threadIdx.x;
    int chunk = (n + 1023) >> 10;
    int beg = tid * chunk;
    int end = beg + chunk; if (end > n) end = n; if (beg > n) beg = n;
    int s = 0;
    for (int i = beg; i < end; ++i) s += cnt[i];
    ssum[tid] = s;
    __syncthreads();
    // Hillis-Steele inclusive scan over the 1024 partials
    for (int off = 1; off < 1024; off <<= 1) {
        int v = (tid >= off) ? ssum[tid - off] : 0;
        __syncthreads();
        ssum[tid] += v;
        __syncthreads();
    }
    int base = (tid == 0) ? 0 : ssum[tid - 1];
    for (int i = beg; i < end; ++i) { rowptr[i] = base; base += cnt[i]; }
    if (tid == 0) rowptr[n] = ssum[1023];
}

__global__ void k_place(const int* __restrict__ ei, const int* __restrict__ rowptr,
                        int* __restrict__ fill, int* __restrict__ eix, int E, int n_nodes) {
    int e = blockIdx.x * blockDim.x + threadIdx.x;
    if (e >= E + n_nodes) return;
    int col = (e < E) ? ei[E + e] : (e - E);
    int pos = atomic_add_i32(&fill[col], 1);
    eix[rowptr[col] + pos] = e;
}

// ---------- stage 7: xw = x @ W  (fp32 WMMA, one 16x16 tile per wave) ----------
__global__ void __launch_bounds__(256) k_gemm_xw(const float* __restrict__ x,
                                                 const float* __restrict__ W,
                                                 float* __restrict__ xw) {
    int wave = threadIdx.x >> 5;          // 0..7 -> column tile
    int lane = threadIdx.x & 31;
    int m0 = blockIdx.x * 16;
    int mr = lane & 15;
    int khalf = lane >> 4;                // 0 or 1
    int kh = khalf << 1;                  // 0 or 2
    int n = (wave << 4) + mr;             // global column
    const float* arow = x + (size_t)(m0 + mr) * FDIM;
    v8f c = {};
    #pragma unroll
    for (int k = 0; k < FDIM; k += 4) {
        v2f a, b;
        a.x = arow[k + kh];
        a.y = arow[k + kh + 1];
        b.x = W[(size_t)(k + kh) * FDIM + n];
        b.y = W[(size_t)(k + kh + 1) * FDIM + n];
        c = __builtin_amdgcn_wmma_f32_16x16x4_f32(false, a, false, b, (short)0, c, false, false);
    }
    #pragma unroll
    for (int v = 0; v < 8; ++v) {
        int row = m0 + v + (khalf << 3);
        xw[(size_t)row * FDIM + n] = c[v];
    }
}

// ---------- stage 8: gather aggregation, one wave per target node ----------
__global__ void k_gather(const int* __restrict__ ei, const float* __restrict__ ew,
                         const float* __restrict__ dinv, const float* __restrict__ xw,
                         const int* __restrict__ rowptr, const int* __restrict__ eix,
                         float* __restrict__ h, int E, int n_nodes) {
    int gid = blockIdx.x * blockDim.x + threadIdx.x;
    int node = gid >> 5;
    int lane = gid & 31;
    if (node >= n_nodes) return;
    int beg = rowptr[node], end = rowptr[node + 1];
    float dc = dinv[node];
    float4 acc = make_float4(0.f, 0.f, 0.f, 0.f);
    for (int idx = beg; idx < end; ++idx) {
        int e = eix[idx];
        int row; float w;
        if (e < E) { row = ei[e]; w = ew[e]; }
        else       { row = e - E; w = 1.0f; }
        float nrm = dinv[row] * w * dc;
        float4 v = ((const float4*)(xw + (size_t)row * FDIM))[lane];
        acc.x += nrm * v.x; acc.y += nrm * v.y;
        acc.z += nrm * v.z; acc.w += nrm * v.w;
    }
    ((float4*)(h + (size_t)node * FDIM))[lane] = acc;
}

// ---------- stage 9: out = relu(h) @ W_lin^T + b_lin  (fp32 WMMA) ----------
__global__ void __launch_bounds__(256) k_gemm_out(const float* __restrict__ h,
                                                  const float* __restrict__ Wlin,
                                                  const float* __restrict__ blin,
                                                  float* __restrict__ out) {
    int wave = threadIdx.x >> 5;
    int lane = threadIdx.x & 31;
    int m0 = blockIdx.x * 16;
    int mr = lane & 15;
    int khalf = lane >> 4;
    int kh = khalf << 1;
    int n = (wave << 4) + mr;
    const float* arow = h + (size_t)(m0 + mr) * FDIM;
    const float* brow = Wlin + (size_t)n * FDIM;   // B[k][n] = Wlin[n][k]
    v8f c = {};
    #pragma unroll
    for (int k = 0; k < FDIM; k += 4) {
        v2f a, b;
        a.x = fmaxf(arow[k + kh], 0.f);
        a.y = fmaxf(arow[k + kh + 1], 0.f);
        b.x = brow[k + kh];
        b.y = brow[k + kh + 1];
        c = __builtin_amdgcn_wmma_f32_16x16x4_f32(false, a, false, b, (short)0, c, false, false);
    }
    float bias = blin[n];
    #pragma unroll
    for (int v = 0; v < 8; ++v) {
        int row = m0 + v + (khalf << 3);
        out[(size_t)row * FDIM + n] = c[v] + bias;
    }
}

extern "C" void kernel_launch(void* const* d_in, const int* in_sizes, int n_in,
                              void* d_out, int out_size, void* d_ws, size_t ws_size,
                              hipStream_t stream) {
    const float* x     = (const float*)d_in[0];
    const int*   ei    = (const int*)  d_in[1];
    const float* ew    = (const float*)d_in[2];
    const float* p     = (const float*)d_in[3];
    const float* w_ih  = (const float*)d_in[4];
    const float* w_hh  = (const float*)d_in[5];
    const float* b_ih  = (const float*)d_in[6];
    const float* b_hh  = (const float*)d_in[7];
    const float* W0    = (const float*)d_in[8];
    const float* Wlin  = (const float*)d_in[9];
    const float* blin  = (const float*)d_in[10];
    float* out = (float*)d_out;

    const int n_nodes = in_sizes[0] / FDIM;   // 50000
    const int E       = in_sizes[1] / 2;      // 800000
    const int nE      = E + n_nodes;          // edges incl. self loops

    // workspace carve-up (all counts multiple of 4 -> float4-safe alignment)
    float* score  = (float*)d_ws;                         // n_nodes
    float* deg    = score  + n_nodes;                     // n_nodes
    float* dinv   = deg    + n_nodes;                     // n_nodes
    float* pinv   = dinv   + n_nodes;                     // 4 (1 used)
    float* topval = pinv   + 4;                           // 128
    int*   perm   = (int*)(topval + FDIM);                // 128
    float* xt     = (float*)(perm + FDIM);                // 128*128
    float* gi     = xt  + FDIM * FDIM;                    // 128*384
    float* gh     = gi  + FDIM * 3 * FDIM;                // 128*384
    float* Wev    = gh  + FDIM * 3 * FDIM;                // 128*128
    float* xw     = Wev + FDIM * FDIM;                    // n_nodes*128
    float* hbuf   = xw  + (size_t)n_nodes * FDIM;         // n_nodes*128
    int*   cnt    = (int*)(hbuf + (size_t)n_nodes * FDIM);// n_nodes
    int*   fill   = cnt  + n_nodes;                       // n_nodes
    int*   rowptr = fill + n_nodes;                       // n_nodes+4
    int*   eix    = rowptr + n_nodes + 4;                 // nE

    // stage 0-2: scoring + top-k
    k_pnorm<<<1, FDIM, 0, stream>>>(p, pinv);
    k_score<<<(n_nodes + 255) / 256, 256, 0, stream>>>(x, p, pinv, score, n_nodes);
    k_topk<<<1, 1024, 0, stream>>>(score, perm, topval, n_nodes);

    // stage 3-5: GRU weight evolution
    k_xtilde<<<(FDIM * FDIM) / 256, 256, 0, stream>>>(x, perm, topval, xt);
    k_gru_mm<<<(FDIM * 3 * FDIM + 255) / 256, 256, 0, stream>>>(xt, W0, w_ih, w_hh, b_ih, b_hh, gi, gh);
    k_gru_combine<<<(FDIM * FDIM) / 256, 256, 0, stream>>>(gi, gh, W0, Wev);

    // stage 6: degrees + CSR-by-target (scratch must be re-zeroed every call)
    k_zero_f32<<<(n_nodes + 255) / 256, 256, 0, stream>>>(deg, n_nodes);
    k_zero_i32<<<(n_nodes + 255) / 256, 256, 0, stream>>>(cnt, n_nodes);
    k_zero_i32<<<(n_nodes + 255) / 256, 256, 0, stream>>>(fill, n_nodes);
    k_deg<<<(nE + 255) / 256, 256, 0, stream>>>(ei, ew, deg, E, n_nodes);
    k_dinv<<<(n_nodes + 255) / 256, 256, 0, stream>>>(deg, dinv, n_nodes);
    k_count<<<(nE + 255) / 256, 256, 0, stream>>>(ei, cnt, E, n_nodes);
    k_scan<<<1, 1024, 0, stream>>>(cnt, rowptr, n_nodes);
    k_place<<<(nE + 255) / 256, 256, 0, stream>>>(ei, rowptr, fill, eix, E, n_nodes);

    // stage 7: xw = x @ Wev  (WMMA)
    k_gemm_xw<<<n_nodes / 16, 256, 0, stream>>>(x, Wev, xw);

    // stage 8: gather aggregation (no fp atomics, h fully written)
    {
        long long threads = (long long)n_nodes * 32;
        k_gather<<<(int)((threads + 255) / 256), 256, 0, stream>>>(ei, ew, dinv, xw, rowptr, eix, hbuf, E, n_nodes);
    }

    // stage 9: out = relu(h) @ Wlin^T + blin  (WMMA)
    k_gemm_out<<<n_nodes / 16, 256, 0, stream>>>(hbuf, Wlin, blin, out);
    (void)n_in; (void)out_size; (void)ws_size;
}